// FilterData_1219770712852
// MI455X (gfx1250) — compile-verified
//
#include <hip/hip_runtime.h>
#include <math.h>

// ---------------------------------------------------------------------------
// Problem constants (from reference): T=50, B=50, C=4096, 2 bands, padlen=49
// Filtfilt+detrend is linear along time -> per band a 50x50 operator M.
// Y[band] = M_band (50x50) @ X (50x204800)  -- done with V_WMMA_F32_16X16X4_F32
// ---------------------------------------------------------------------------
#define T_LEN   50
#define CCH     4096
#define NCOLS   204800              // 50*4096 flattened (b,c)
#define LDA     52                  // K padded 50 -> 52 (13 k-steps of 4)
#define MPAD    64                  // M padded 50 -> 64 (4 tiles of 16)
#define PADLEN  49
#define EXTLEN  148                 // 50 + 2*49
#define OUT_PER_BAND 10240000       // 50*204800

typedef float v2f __attribute__((ext_vector_type(2)));
typedef float v8f __attribute__((ext_vector_type(8)));

// ---------------------- complex double helpers (filter design) -------------
struct cdbl { double re, im; };
__device__ __forceinline__ cdbl cadd(cdbl a, cdbl b){ return {a.re+b.re, a.im+b.im}; }
__device__ __forceinline__ cdbl csub(cdbl a, cdbl b){ return {a.re-b.re, a.im-b.im}; }
__device__ __forceinline__ cdbl cmul(cdbl a, cdbl b){ return {a.re*b.re-a.im*b.im, a.re*b.im+a.im*b.re}; }
__device__ __forceinline__ cdbl cdivc(cdbl a, cdbl b){
  double d = b.re*b.re + b.im*b.im;
  return {(a.re*b.re+a.im*b.im)/d, (a.im*b.re-a.re*b.im)/d};
}
__device__ __forceinline__ cdbl csqrtc(cdbl z){            // principal branch
  double r  = sqrt(z.re*z.re + z.im*z.im);
  double re = sqrt(fmax(0.0, 0.5*(r + z.re)));
  double im = sqrt(fmax(0.0, 0.5*(r - z.re)));
  if (z.im < 0.0) im = -im;
  return {re, im};
}

// Butterworth bandpass (n=2) -> 2 SOS sections, mirrors reference math exactly.
__device__ void butter_bp_sos(double w1, double w2, double sos[2][6]) {
  const double PI = 3.14159265358979323846;
  double warped1 = 4.0 * tan(PI * w1 * 0.5);   // 2*fs*tan(pi*w/fs), fs=2
  double warped2 = 4.0 * tan(PI * w2 * 0.5);
  double bw = warped2 - warped1;
  double wo = sqrt(warped1 * warped2);
  cdbl pbp[4];
  #pragma unroll
  for (int k = 0; k < 2; ++k) {
    double ang = PI * (2.0*(k+1) - 1.0) / 4.0;          // (2k-1)pi/(2n), n=2
    cdbl p   = { -cos(ang), -sin(ang) };                 // -exp(i*ang)
    cdbl plp = { p.re*bw*0.5, p.im*bw*0.5 };
    cdbl d   = csqrtc(csub(cmul(plp, plp), {wo*wo, 0.0}));
    pbp[k]   = cadd(plp, d);
    pbp[k+2] = csub(plp, d);
  }
  cdbl prod = {1.0, 0.0};
  #pragma unroll
  for (int k = 0; k < 4; ++k) prod = cmul(prod, csub({4.0,0.0}, pbp[k]));
  double gain = bw*bw * 16.0 / prod.re;                  // bw^n * fs2^n / Re(prod)
  int nset = 0;
  for (int k = 0; k < 4 && nset < 2; ++k) {
    cdbl pd = cdivc(cadd({4.0,0.0}, pbp[k]), csub({4.0,0.0}, pbp[k]));
    if (pd.im > 0.0) {
      double g = (nset == 0) ? gain : 1.0;
      sos[nset][0] = g;   sos[nset][1] = 0.0;            sos[nset][2] = -g;
      sos[nset][3] = 1.0; sos[nset][4] = -2.0*pd.re;     sos[nset][5] = pd.re*pd.re + pd.im*pd.im;
      ++nset;
    }
  }
}

__device__ void sosfilt_zi(const double sos[2][6], double zi[2][2]) {
  double scale = 1.0;
  #pragma unroll
  for (int s = 0; s < 2; ++s) {
    double b0=sos[s][0], b1=sos[s][1], b2=sos[s][2], a1=sos[s][4], a2=sos[s][5];
    double B0 = b1 - a1*b0, B1 = b2 - a2*b0, det = 1.0 + a1 + a2;
    zi[s][0] = scale * (B0 + B1) / det;
    zi[s][1] = scale * ((1.0 + a1)*B1 - a2*B0) / det;
    scale *= (b0 + b1 + b2) / (1.0 + a1 + a2);
  }
}

// ---------------------------------------------------------------------------
// Kernel 1: build M_band columns (filtfilt of a time basis vector + detrend),
// stored as A[band][64][52] f32, zero padded. 104 tiny sequential f64 tasks.
// ---------------------------------------------------------------------------
__global__ void build_M_kernel(float* __restrict__ A) {
  int tid = blockIdx.x * blockDim.x + threadIdx.x;
  if (tid >= 2 * LDA) return;
  int band = tid / LDA;
  int j    = tid % LDA;
  float* Acol = A + band * MPAD * LDA + j;
  if (j >= T_LEN) {                       // K padding columns
    for (int r = 0; r < MPAD; ++r) Acol[r * LDA] = 0.0f;
    return;
  }
  double w1 = (band == 0) ? 0.05 : 0.20;
  double w2 = (band == 0) ? 0.15 : 0.40;
  double sos[2][6], zi[2][2];
  butter_bp_sos(w1, w2, sos);
  sosfilt_zi(sos, zi);

  double e[EXTLEN];
  // odd extension of basis vector e_j
  for (int k = 0; k < PADLEN; ++k)
    e[k] = 2.0*((j==0)?1.0:0.0) - (((PADLEN-k)==j)?1.0:0.0);
  for (int i = 0; i < T_LEN; ++i)
    e[PADLEN+i] = (i==j) ? 1.0 : 0.0;
  for (int k = 0; k < PADLEN; ++k)
    e[PADLEN+T_LEN+k] = 2.0*((j==T_LEN-1)?1.0:0.0) - (((T_LEN-2-k)==j)?1.0:0.0);

  for (int pass = 0; pass < 2; ++pass) {
    double x0 = e[0];                     // zi scaled by first sample (pre-filter)
    for (int s = 0; s < 2; ++s) {
      double b0=sos[s][0], b1=sos[s][1], b2=sos[s][2], a1=sos[s][4], a2=sos[s][5];
      double z0 = zi[s][0]*x0, z1 = zi[s][1]*x0;
      for (int t = 0; t < EXTLEN; ++t) {  // direct form II transposed
        double xt = e[t];
        double yt = b0*xt + z0;
        z0 = b1*xt + z1 - a1*yt;
        z1 = b2*xt - a2*yt;
        e[t] = yt;
      }
    }
    // reverse (between passes, and back to forward order after pass 2)
    for (int i = 0; i < EXTLEN/2; ++i) { double t0=e[i]; e[i]=e[EXTLEN-1-i]; e[EXTLEN-1-i]=t0; }
  }
  // slice + fold detrend (subtract column mean) into M
  double mean = 0.0;
  for (int t = 0; t < T_LEN; ++t) mean += e[PADLEN+t];
  mean /= (double)T_LEN;
  for (int r = 0; r < MPAD; ++r)
    Acol[r * LDA] = (r < T_LEN) ? (float)(e[PADLEN+r] - mean) : 0.0f;
}

// ---------------------------------------------------------------------------
// Kernel 2: Y[band] = M_band @ X via V_WMMA_F32_16X16X4_F32.
// One wave = one 16-column N-tile, all 4 M-tiles (full time axis), both bands
// sharing the B fragment. Epilogue computes per-column sum/sumsq for zscore.
// ---------------------------------------------------------------------------
__global__ __launch_bounds__(256) void gemm_filt_kernel(
    const float* __restrict__ X, const float* __restrict__ A,
    float* __restrict__ out, float* __restrict__ stat_m, float* __restrict__ stat_s) {
  const int lane  = threadIdx.x & 31;
  const int wave  = threadIdx.x >> 5;
  const int col   = lane & 15;       // N within tile (also A row within M-tile)
  const int khalf = lane >> 4;       // 0: K+0/K+1, 1: K+2/K+3
  const long n0   = ((long)blockIdx.x * 8 + wave) * 16;

  v8f acc[2][4];
  #pragma unroll
  for (int b = 0; b < 2; ++b)
    #pragma unroll
    for (int m = 0; m < 4; ++m) acc[b][m] = {};

  const float* Xcol = X + n0 + col;
  for (int k = 0; k < LDA; k += 4) {
    // B fragment (branchless K-tail: keep EXEC all-ones for WMMA)
    int r0 = k + 2*khalf;
    int r1 = r0 + 1;
    v2f bfrag;
    bfrag.x = Xcol[(long)(r0 < T_LEN ? r0 : T_LEN-1) * NCOLS] * (r0 < T_LEN ? 1.0f : 0.0f);
    bfrag.y = Xcol[(long)(r1 < T_LEN ? r1 : T_LEN-1) * NCOLS] * (r1 < T_LEN ? 1.0f : 0.0f);
    #pragma unroll
    for (int band = 0; band < 2; ++band) {
      const float* Ab = A + band * MPAD * LDA;
      #pragma unroll
      for (int mt = 0; mt < 4; ++mt) {
        // A 16x4 layout: lanes 0-15 rows M=0..15 (K=k,k+1), lanes 16-31 same rows (K=k+2,k+3)
        v2f afrag = *(const v2f*)(Ab + (mt*16 + col) * LDA + r0);
        acc[band][mt] = __builtin_amdgcn_wmma_f32_16x16x4_f32(
            false, afrag, false, bfrag, (short)0, acc[band][mt], false, false);
      }
    }
  }

  // Epilogue: per-column time stats (rows 50..63 are exactly 0 -> harmless)
  #pragma unroll
  for (int band = 0; band < 2; ++band) {
    float sum = 0.0f, sq = 0.0f;
    #pragma unroll
    for (int mt = 0; mt < 4; ++mt)
      #pragma unroll
      for (int r = 0; r < 8; ++r) { float v = acc[band][mt][r]; sum += v; sq += v*v; }
    sum += __shfl_xor(sum, 16, 32);   // fold lane halves (same column)
    sq  += __shfl_xor(sq,  16, 32);
    float m   = sum / (float)T_LEN;
    float var = (sq - sum*sum/(float)T_LEN) / (float)(T_LEN - 1); // ddof=1
    float s   = sqrtf(var > 0.0f ? var : 0.0f);
    if (lane < 16) {
      stat_m[band*NCOLS + n0 + lane] = m;
      stat_s[band*NCOLS + n0 + lane] = s;
    }
    // store detrended filtered data (C/D layout: M = mt*16 + 8*khalf + r)
    float* Ob = out + (long)band * OUT_PER_BAND + n0 + col;
    #pragma unroll
    for (int mt = 0; mt < 4; ++mt)
      #pragma unroll
      for (int r = 0; r < 8; ++r) {
        int t = mt*16 + 8*khalf + r;
        if (t < T_LEN) Ob[(long)t * NCOLS] = acc[band][mt][r];
      }
  }
}

// ---------------------------------------------------------------------------
// Kernel 3: out[t,b,c] = (y[t,b,c] - m[t,c]) / s[t,c]  (reference's transposed
// broadcast since T==B). In-place, float4 vectorized.
// ---------------------------------------------------------------------------
__global__ __launch_bounds__(256) void norm_kernel(
    float* __restrict__ out, const float* __restrict__ stat_m,
    const float* __restrict__ stat_s) {
  long idx4 = (long)blockIdx.x * blockDim.x + threadIdx.x;
  long idx  = idx4 * 4;
  if (idx >= 2L * OUT_PER_BAND) return;
  long band = idx / OUT_PER_BAND;
  long rem  = idx % OUT_PER_BAND;
  long t    = rem / NCOLS;
  long c    = rem % CCH;                      // (n % 4096); 4-aligned
  long si   = band * NCOLS + t * CCH + c;     // stats of series (b=t, c)
  float4 v  = *reinterpret_cast<float4*>(out + idx);
  float4 mm = *reinterpret_cast<const float4*>(stat_m + si);
  float4 ss = *reinterpret_cast<const float4*>(stat_s + si);
  v.x = (v.x - mm.x) / ss.x;
  v.y = (v.y - mm.y) / ss.y;
  v.z = (v.z - mm.z) / ss.z;
  v.w = (v.w - mm.w) / ss.w;
  *reinterpret_cast<float4*>(out + idx) = v;
}

// ---------------------------------------------------------------------------
extern "C" void kernel_launch(void* const* d_in, const int* in_sizes, int n_in,
                              void* d_out, int out_size, void* d_ws, size_t ws_size,
                              hipStream_t stream) {
  const float* X  = (const float*)d_in[0];      // [50, 50, 4096] f32
  float* out      = (float*)d_out;              // [2, 50, 50, 4096] f32
  float* A        = (float*)d_ws;                                   // 26.6 KB
  float* stat_m   = (float*)((char*)d_ws + 32768);                  // 1.6 MB
  float* stat_s   = (float*)((char*)d_ws + 32768 + 2*NCOLS*sizeof(float));

  hipLaunchKernelGGL(build_M_kernel, dim3(1), dim3(128), 0, stream, A);
  hipLaunchKernelGGL(gemm_filt_kernel, dim3(NCOLS/16/8), dim3(256), 0, stream,
                     X, A, out, stat_m, stat_s);
  hipLaunchKernelGGL(norm_kernel, dim3((2*OUT_PER_BAND/4 + 255)/256), dim3(256),
                     0, stream, out, stat_m, stat_s);
}